// GATModel_33938831573709
// MI455X (gfx1250) — compile-verified
//
#include <hip/hip_runtime.h>
#include <hip/hip_bf16.h>

#define N_NODES   100000
#define N_EDGES   1600000
#define D_IN      256
#define D_HID     128
#define N_CLASSES 64
#define NEG_SLOPE 0.2f

typedef __attribute__((ext_vector_type(2))) float v2f;
typedef __attribute__((ext_vector_type(8))) float v8f;

// ---------------------------------------------------------------------------
// GEMM1: feat[N, 128] = X[N, 256] @ W[256, 128]   (fp32 WMMA 16x16x4)
// One wave computes a 16x32 output strip (2 independent accumulators -> 2-way
// WMMA ILP, A fragment reused for both col tiles). 8 waves / 256-thread block.
// Strips: 6250 row tiles x 4 col pairs = 25000 -> 3125 blocks (exact).
// ---------------------------------------------------------------------------
__global__ __launch_bounds__(256) void k_gemm1(const float* __restrict__ X,
                                               const float* __restrict__ W,
                                               float* __restrict__ feat) {
    const int wave = threadIdx.x >> 5;
    const int lane = threadIdx.x & 31;
    const int strip = blockIdx.x * 8 + wave;      // 0..24999
    const int rowTile = strip >> 2;               // /4 col pairs
    const int colPair = strip & 3;
    const int row0 = rowTile * 16;
    const int col0 = colPair * 32;
    const int l16 = lane & 15;
    const int hi  = lane >> 4;                    // 0 for lanes 0-15, 1 for 16-31

    v8f acc0 = {};
    v8f acc1 = {};
    const float* __restrict__ xrow = X + (size_t)(row0 + l16) * D_IN;

    for (int k0 = 0; k0 < D_IN; k0 += 4) {
        // A 16x4 fp32: lanes 0-15 -> K={k0,k0+1}; lanes 16-31 -> K={k0+2,k0+3}
        const int ka = k0 + 2 * hi;
        v2f a;
        a.x = xrow[ka];
        a.y = xrow[ka + 1];
        // B 4x16 fp32 fragments for two adjacent col tiles
        const float* __restrict__ w0 = W + (size_t)ka * D_HID + col0 + l16;
        const float* __restrict__ w1 = W + (size_t)(ka + 1) * D_HID + col0 + l16;
        v2f b0, b1;
        b0.x = w0[0];   b0.y = w1[0];
        b1.x = w0[16];  b1.y = w1[16];
        acc0 = __builtin_amdgcn_wmma_f32_16x16x4_f32(
            false, a, false, b0, (short)0, acc0, false, false);
        acc1 = __builtin_amdgcn_wmma_f32_16x16x4_f32(
            false, a, false, b1, (short)0, acc1, false, false);
    }
    // C/D 16x16 fp32: VGPR r -> M = r + 8*hi, N = col + l16
    #pragma unroll
    for (int r = 0; r < 8; ++r) {
        float* __restrict__ orow = feat + (size_t)(row0 + r + 8 * hi) * D_HID + col0 + l16;
        orow[0]  = acc0[r];
        orow[16] = acc1[r];
    }
}

// ---------------------------------------------------------------------------
// GEMM2: out[N, 64] = h[N, 128] @ fc_W[128, 64] + fc_b   (fp32 WMMA 16x16x4)
// One wave computes a full 16x64 output strip (4 accumulators -> 4-way WMMA
// ILP, A loaded once per k-step). 6250 strips -> 782 blocks, wave-uniform tail
// guard (keeps EXEC all-ones inside active waves, as WMMA requires).
// ---------------------------------------------------------------------------
__global__ __launch_bounds__(256) void k_gemm2(const float* __restrict__ H,
                                               const float* __restrict__ FW,
                                               const float* __restrict__ FB,
                                               float* __restrict__ out) {
    const int wave = threadIdx.x >> 5;
    const int lane = threadIdx.x & 31;
    const int rowTile = blockIdx.x * 8 + wave;    // 0..6255 (guard below)
    if (rowTile >= N_NODES / 16) return;          // wave-uniform
    const int row0 = rowTile * 16;
    const int l16 = lane & 15;
    const int hi  = lane >> 4;

    v8f acc0 = {}, acc1 = {}, acc2 = {}, acc3 = {};
    const float* __restrict__ hrow = H + (size_t)(row0 + l16) * D_HID;

    for (int k0 = 0; k0 < D_HID; k0 += 4) {
        const int ka = k0 + 2 * hi;
        v2f a;
        a.x = hrow[ka];
        a.y = hrow[ka + 1];
        const float* __restrict__ w0 = FW + (size_t)ka * N_CLASSES + l16;
        const float* __restrict__ w1 = FW + (size_t)(ka + 1) * N_CLASSES + l16;
        v2f b0, b1, b2, b3;
        b0.x = w0[0];   b0.y = w1[0];
        b1.x = w0[16];  b1.y = w1[16];
        b2.x = w0[32];  b2.y = w1[32];
        b3.x = w0[48];  b3.y = w1[48];
        acc0 = __builtin_amdgcn_wmma_f32_16x16x4_f32(
            false, a, false, b0, (short)0, acc0, false, false);
        acc1 = __builtin_amdgcn_wmma_f32_16x16x4_f32(
            false, a, false, b1, (short)0, acc1, false, false);
        acc2 = __builtin_amdgcn_wmma_f32_16x16x4_f32(
            false, a, false, b2, (short)0, acc2, false, false);
        acc3 = __builtin_amdgcn_wmma_f32_16x16x4_f32(
            false, a, false, b3, (short)0, acc3, false, false);
    }
    const float bias0 = FB[l16];
    const float bias1 = FB[l16 + 16];
    const float bias2 = FB[l16 + 32];
    const float bias3 = FB[l16 + 48];
    #pragma unroll
    for (int r = 0; r < 8; ++r) {
        float* __restrict__ orow = out + (size_t)(row0 + r + 8 * hi) * N_CLASSES + l16;
        orow[0]  = acc0[r] + bias0;
        orow[16] = acc1[r] + bias1;
        orow[32] = acc2[r] + bias2;
        orow[48] = acc3[r] + bias3;
    }
}

// ---------------------------------------------------------------------------
// Attention scalars: el[i] = feat[i] . attn_l ; er[i] = feat[i] . attn_r
// One wave per node; shfl reduction (wave32).
// ---------------------------------------------------------------------------
__global__ __launch_bounds__(256) void k_scores(const float* __restrict__ feat,
                                                const float* __restrict__ al,
                                                const float* __restrict__ ar,
                                                float* __restrict__ el,
                                                float* __restrict__ er) {
    const int wave = threadIdx.x >> 5;
    const int lane = threadIdx.x & 31;
    const int node = blockIdx.x * 8 + wave;
    if (node >= N_NODES) return;
    const float* __restrict__ f = feat + (size_t)node * D_HID;
    float sl = 0.f, sr = 0.f;
    #pragma unroll
    for (int j = 0; j < D_HID; j += 32) {
        float v = f[j + lane];
        sl += v * al[j + lane];
        sr += v * ar[j + lane];
    }
    #pragma unroll
    for (int off = 16; off > 0; off >>= 1) {
        sl += __shfl_down(sl, off, 32);
        sr += __shfl_down(sr, off, 32);
    }
    if (lane == 0) { el[node] = sl; er[node] = sr; }
}

// Monotonic float<->uint encoding for atomicMax-based segment max.
__device__ __forceinline__ unsigned enc_f32(float f) {
    unsigned b = __float_as_uint(f);
    return (b & 0x80000000u) ? ~b : (b | 0x80000000u);
}
__device__ __forceinline__ float dec_f32(unsigned u) {
    unsigned b = (u & 0x80000000u) ? (u & 0x7FFFFFFFu) : ~u;
    return __uint_as_float(b);
}

// ---------------------------------------------------------------------------
// Edge pass 1: e = leaky_relu(el[src] + er[dst]); segment max into mEnc[dst]
// ---------------------------------------------------------------------------
__global__ __launch_bounds__(256) void k_edge1(const float* __restrict__ el,
                                               const float* __restrict__ er,
                                               const int* __restrict__ src,
                                               const int* __restrict__ dst,
                                               float* __restrict__ e,
                                               unsigned* __restrict__ mEnc) {
    const int i = blockIdx.x * 256 + threadIdx.x;
    if (i >= N_EDGES) return;
    float v = el[src[i]] + er[dst[i]];
    v = (v > 0.f) ? v : NEG_SLOPE * v;
    e[i] = v;
    atomicMax(&mEnc[dst[i]], enc_f32(v));
}

// ---------------------------------------------------------------------------
// Edge pass 2: ex = exp(e - m[dst]); denom[dst] += ex   (overwrite e with ex)
// ---------------------------------------------------------------------------
__global__ __launch_bounds__(256) void k_edge2(const int* __restrict__ dst,
                                               const unsigned* __restrict__ mEnc,
                                               float* __restrict__ e,
                                               float* __restrict__ denom) {
    const int i = blockIdx.x * 256 + threadIdx.x;
    if (i >= N_EDGES) return;
    const int d = dst[i];
    const float ex = __expf(e[i] - dec_f32(mEnc[d]));
    e[i] = ex;
    atomicAdd(&denom[d], ex);
}

// ---------------------------------------------------------------------------
// Edge pass 3: a = ex / denom[dst]; h[dst, :] += a * feat[src, :]
// 128 threads per edge (one per hidden channel), 2 edges per 256-thread block.
// feat and h both fit in the 192 MB L2 -> gathers and f32 atomic adds run at
// L2 bandwidth. This is the dominant cost (~1.6 GB of L2 traffic).
// ---------------------------------------------------------------------------
__global__ __launch_bounds__(256) void k_edge3(const int* __restrict__ src,
                                               const int* __restrict__ dst,
                                               const float* __restrict__ ex,
                                               const float* __restrict__ denom,
                                               const float* __restrict__ feat,
                                               float* __restrict__ h) {
    const int eIdx = blockIdx.x * 2 + (threadIdx.x >> 7);
    const int t    = threadIdx.x & 127;
    if (eIdx >= N_EDGES) return;
    const int d = dst[eIdx];
    const int s = src[eIdx];
    const float a = ex[eIdx] / denom[d];
    atomicAdd(&h[(size_t)d * D_HID + t], a * feat[(size_t)s * D_HID + t]);
}

// ---------------------------------------------------------------------------
// Launch
// inputs: 0 node_features [N,256], 1 W [256,128], 2 attn_l [128], 3 attn_r [128],
//         4 fc_W [128,64], 5 fc_b [64], 6 src [E], 7 dst [E]
// ---------------------------------------------------------------------------
extern "C" void kernel_launch(void* const* d_in, const int* in_sizes, int n_in,
                              void* d_out, int out_size, void* d_ws, size_t ws_size,
                              hipStream_t stream) {
    const float* X  = (const float*)d_in[0];
    const float* W  = (const float*)d_in[1];
    const float* al = (const float*)d_in[2];
    const float* ar = (const float*)d_in[3];
    const float* FW = (const float*)d_in[4];
    const float* FB = (const float*)d_in[5];
    const int*  src = (const int*)d_in[6];
    const int*  dst = (const int*)d_in[7];
    float* out = (float*)d_out;

    // Workspace carve-out (256-byte aligned chunks)
    char* ws = (char*)d_ws;
    size_t off = 0;
    auto carve = [&](size_t bytes) {
        void* p = ws + off;
        off += (bytes + 255) & ~(size_t)255;
        return p;
    };
    float*    feat  = (float*)   carve((size_t)N_NODES * D_HID * sizeof(float)); // 51.2 MB
    float*    h     = (float*)   carve((size_t)N_NODES * D_HID * sizeof(float)); // 51.2 MB
    float*    el    = (float*)   carve((size_t)N_NODES * sizeof(float));
    float*    er    = (float*)   carve((size_t)N_NODES * sizeof(float));
    unsigned* mEnc  = (unsigned*)carve((size_t)N_NODES * sizeof(unsigned));
    float*    denom = (float*)   carve((size_t)N_NODES * sizeof(float));
    float*    e     = (float*)   carve((size_t)N_EDGES * sizeof(float));         // 6.4 MB

    // Zero accumulators (graph-capture safe)
    hipMemsetAsync(h,     0, (size_t)N_NODES * D_HID * sizeof(float), stream);
    hipMemsetAsync(mEnc,  0, (size_t)N_NODES * sizeof(unsigned), stream); // 0 == encoded minimum
    hipMemsetAsync(denom, 0, (size_t)N_NODES * sizeof(float), stream);

    // 1) feat = X @ W            (25000 strips / 8 waves = 3125 blocks)
    k_gemm1<<<3125, 256, 0, stream>>>(X, W, feat);
    // 2) el, er                  (100000 nodes / 8 waves = 12500 blocks)
    k_scores<<<12500, 256, 0, stream>>>(feat, al, ar, el, er);
    // 3) edge score + segment max
    k_edge1<<<(N_EDGES + 255) / 256, 256, 0, stream>>>(el, er, src, dst, e, mEnc);
    // 4) exp + segment sum
    k_edge2<<<(N_EDGES + 255) / 256, 256, 0, stream>>>(dst, mEnc, e, denom);
    // 5) normalized scatter-add aggregation (2 edges / block)
    k_edge3<<<N_EDGES / 2, 256, 0, stream>>>(src, dst, e, denom, feat, h);
    // 6) out = h @ fc_W + fc_b   (6250 strips / 8 waves = 782 blocks, tail guard)
    k_gemm2<<<782, 256, 0, stream>>>(h, FW, FB, out);
}